// RRCP_prediction_54949811585479
// MI455X (gfx1250) — compile-verified
//
#include <hip/hip_runtime.h>
#include <stdint.h>

// Problem constants from the reference.
#define B 256
#define N 100   // stored retrieval dim (inputs)
#define D 768
#define R 80    // RETRIEVAL_NUM (used slice)

// Flat output layout (reference tuple order, all f32):
//   vis_packed [B,R,D], txt_packed [B,R,D], text_mask [B,R+1],
//   img_mask [B,R+1], rr_mod [B,R], labels [B,R]
static constexpr size_t VIS_OFF = 0;
static constexpr size_t TXT_OFF = (size_t)B * R * D;                 // 15,728,640
static constexpr size_t TM_OFF  = 2 * (size_t)B * R * D;             // 31,457,280
static constexpr size_t IM_OFF  = TM_OFF + (size_t)B * (R + 1);      // 31,478,016
static constexpr size_t RR_OFF  = IM_OFF + (size_t)B * (R + 1);      // 31,498,752
static constexpr size_t LAB_OFF = RR_OFF + (size_t)B * R;            // 31,519,232

// CDNA5 async global<->LDS DMA path (ASYNCcnt-tracked). Guarded so the file
// compiles on toolchains where the builtins are absent.
#if defined(__gfx1250__) && \
    __has_builtin(__builtin_amdgcn_global_load_async_to_lds_b128) && \
    __has_builtin(__builtin_amdgcn_global_store_async_from_lds_b128)
#define USE_ASYNC_LDS 1
#else
#define USE_ASYNC_LDS 0
#endif

// Exact pointee type from the compiler diagnostic:
//   "__attribute__((__vector_size__(4 * sizeof(int)))) int __device__ *"
// -> GCC-style int4 vector, addrspace(1) ("__device__") for the global side,
//    addrspace(3) for the LDS side.
typedef int v4i __attribute__((vector_size(4 * sizeof(int))));
typedef __attribute__((address_space(1))) v4i* gv4p;   // global int4*
typedef __attribute__((address_space(3))) v4i* lv4p;   // LDS int4*

__device__ __forceinline__ void wait_asynccnt0() {
#if __has_builtin(__builtin_amdgcn_s_wait_asynccnt)
    __builtin_amdgcn_s_wait_asynccnt(0);
#else
    asm volatile("s_wait_asynccnt 0" ::: "memory");
#endif
}

// ---------------------------------------------------------------------------
// Kernel 1: per-batch metadata. One block per batch row.
//  - binary = rr > 0.5 (strict), counts = sum(binary)
//  - stable compaction permutation perm[b][k] = k-th index with binary==1
//  - rr_mod = (rr < 0.5 ? 0 : rr); if whole row < 0.5 then rr_mod[0] = 1
//  - labels passthrough; text_mask / img_mask
// ---------------------------------------------------------------------------
__global__ void __launch_bounds__(128)
meta_kernel(const float* __restrict__ rrcp,
            const float* __restrict__ labels_in,
            float* __restrict__ out,
            int* __restrict__ counts,
            int* __restrict__ perm)
{
    const int b = blockIdx.x;
    const int t = threadIdx.x;

    __shared__ float s_rr[R];
    __shared__ int   s_perm[R];
    __shared__ int   s_cnt;
    __shared__ int   s_zero;

    if (t < R) s_rr[t] = rrcp[b * N + t];
    __syncthreads();

    if (t == 0) {
        int c = 0;
        int zr = 1;                       // "all rr_mod == 0" <=> no rr >= 0.5
        for (int j = 0; j < R; ++j) {
            float v = s_rr[j];
            if (v > 0.5f) s_perm[c++] = j;   // stable: original order preserved
            if (!(v < 0.5f)) zr = 0;         // v >= 0.5 -> rr_mod[j] = v != 0
        }
        s_cnt = c;
        s_zero = zr;
        counts[b] = c;
    }
    __syncthreads();

    const int c = s_cnt;
    if (t < R) {
        perm[b * R + t] = (t < c) ? s_perm[t] : 0;

        float v = s_rr[t];
        float m = (v < 0.5f) ? 0.0f : v;
        if (t == 0 && s_zero) m = 1.0f;
        out[RR_OFF  + (size_t)b * R + t] = m;
        out[LAB_OFF + (size_t)b * R + t] = labels_in[b * N + t];
    }
    if (t < R + 1) {
        float tm = (t <= c) ? 1.0f : 0.0f;
        out[TM_OFF + (size_t)b * (R + 1) + t] = tm;
        out[IM_OFF + (size_t)b * (R + 1) + t] = (b == B - 1) ? tm : 1.0f;
    }
}

// ---------------------------------------------------------------------------
// Kernel 2: bulk gather. One block per (batch, output slot); each block moves
// one 3KB vis row + one 3KB txt row (or writes zeros past counts[b]).
// 192 threads x float4 = 768 floats per row, fully coalesced B128 traffic.
// On gfx1250 the valid-row path streams through LDS with the async DMA
// instructions: global_load_async_to_lds_b128 -> s_wait_asynccnt 0 ->
// global_store_async_from_lds_b128 (no VGPR round trip, ASYNCcnt-tracked).
// ---------------------------------------------------------------------------
__global__ void __launch_bounds__(192)
gather_kernel(const float* __restrict__ visIn,
              const float* __restrict__ txtIn,
              const int* __restrict__ counts,
              const int* __restrict__ perm,
              float* __restrict__ out)
{
#if USE_ASYNC_LDS
    __shared__ float lbuf[2 * D];  // 6KB staging: [0,D) vis, [D,2D) txt
#endif
    const int bid  = blockIdx.x;
    const int b    = bid / R;
    const int slot = bid - b * R;
    const int t    = threadIdx.x;          // 0..191
    const int cnt  = counts[b];

    float4* visO = (float4*)(out + VIS_OFF + (size_t)(b * R + slot) * D);
    float4* txtO = (float4*)(out + TXT_OFF + (size_t)(b * R + slot) * D);

    if (slot < cnt) {
        const int src = perm[b * R + slot];
        const float4* visI = (const float4*)(visIn + (size_t)(b * N + src) * D);
        const float4* txtI = (const float4*)(txtIn + (size_t)(b * N + src) * D);
#if USE_ASYNC_LDS
        __builtin_amdgcn_global_load_async_to_lds_b128(
            (gv4p)(visI + t), (lv4p)&lbuf[4 * t], 0, 0);
        __builtin_amdgcn_global_load_async_to_lds_b128(
            (gv4p)(txtI + t), (lv4p)&lbuf[D + 4 * t], 0, 0);
        wait_asynccnt0();   // per-wave: each lane stores the LDS it loaded
        __builtin_amdgcn_global_store_async_from_lds_b128(
            (gv4p)(visO + t), (lv4p)&lbuf[4 * t], 0, 0);
        __builtin_amdgcn_global_store_async_from_lds_b128(
            (gv4p)(txtO + t), (lv4p)&lbuf[D + 4 * t], 0, 0);
        // S_ENDPGM implicitly waits for ASYNCcnt==0.
#else
        visO[t] = visI[t];
        txtO[t] = txtI[t];
#endif
    } else {
        const float4 z = make_float4(0.f, 0.f, 0.f, 0.f);
        visO[t] = z;
        txtO[t] = z;
    }
}

// ---------------------------------------------------------------------------
// Inputs (setup_inputs order):
//  0: mean_pooling_vec (unused)   1: merge_text_vec (unused)
//  2: retrieved_visual_feature_embedding_cls (B,N,1,D)
//  3: retrieved_textual_feature_embedding    (B,N,1,D)
//  4: retrieved_label_list (B,N)  5: RRCP (B,N)
// ---------------------------------------------------------------------------
extern "C" void kernel_launch(void* const* d_in, const int* in_sizes, int n_in,
                              void* d_out, int out_size, void* d_ws, size_t ws_size,
                              hipStream_t stream) {
    (void)in_sizes; (void)n_in; (void)out_size; (void)ws_size;

    const float* vis = (const float*)d_in[2];
    const float* txt = (const float*)d_in[3];
    const float* lab = (const float*)d_in[4];
    const float* rr  = (const float*)d_in[5];
    float* out = (float*)d_out;

    int* counts = (int*)d_ws;        // B ints
    int* perm   = counts + B;        // B*R ints (written every call; poison-safe)

    meta_kernel<<<B, 128, 0, stream>>>(rr, lab, out, counts, perm);
    gather_kernel<<<B * R, 192, 0, stream>>>(vis, txt, counts, perm, out);
}